// FreeNet_2319282340033
// MI455X (gfx1250) — compile-verified
//
#include <hip/hip_runtime.h>

typedef float v2f __attribute__((ext_vector_type(2)));
typedef float v8f __attribute__((ext_vector_type(8)));

#define B_TOT 2048
#define D_IN  1024
#define H     384
#define D_OUT 1024
#define BT    16     // batch rows per workgroup
#define HS    388    // LDS row stride (384+4) -> conflict-free ds_load_b64

// wmma f32 16x16x4: D = A(16x4) * B(4x16) + C
// A per lane: row = lane%16, cols k0+2*(lane/16) .. +1  (v2f)
// B per lane: col = lane%16 (=j), rows k0+2*(lane/16) .. +1 (v2f) == W[j][k] pair
// C/D per lane: col N = lane%16, row M = v + 8*(lane/16) for VGPR v
static __device__ __forceinline__ v8f wmma_f32(v2f a, v2f b, v8f c) {
    return __builtin_amdgcn_wmma_f32_16x16x4_f32(
        /*neg_a=*/false, a, /*neg_b=*/false, b,
        /*c_mod=*/(short)0, c, /*reuse_a=*/false, /*reuse_b=*/false);
}

__global__ __launch_bounds__(256) void freenet_kernel(
    const float* __restrict__ x,      // [B, D_IN]
    const float* __restrict__ Wi,     // [H, D_IN]
    const float* __restrict__ bi,     // [H]
    const float* __restrict__ W_pad,  // [H-1, H, H] (pre-masked: zeros outside triangle)
    const float* __restrict__ b_pad,  // [H-1, H]   (pre-masked)
    const float* __restrict__ Wo,     // [D_OUT, H]
    const float* __restrict__ bo,     // [D_OUT]
    float* __restrict__ out)          // [B, D_OUT]
{
    __shared__ float hs[BT][HS];

    const int tid = threadIdx.x;
    const int ln  = tid & 15;         // lane % 16
    const int hi  = (tid >> 4) & 1;   // lane / 16 within wave32
    const int wv  = tid >> 5;         // wave id 0..7
    const int rowbase = blockIdx.x * BT;

    // ================= input layer: h0 = sqrelu(x @ Wi^T + bi) =================
    {
        const float* xr = x + (size_t)(rowbase + ln) * D_IN + 2 * hi;
        const int j0_0 = wv * 16, j0_1 = j0_0 + 128, j0_2 = j0_0 + 256;
        const float* w0 = Wi + (size_t)(j0_0 + ln) * D_IN + 2 * hi;
        const float* w1 = Wi + (size_t)(j0_1 + ln) * D_IN + 2 * hi;
        const float* w2 = Wi + (size_t)(j0_2 + ln) * D_IN + 2 * hi;
        const float bv0 = bi[j0_0 + ln], bv1 = bi[j0_1 + ln], bv2 = bi[j0_2 + ln];
        v8f c0, c1, c2;
        #pragma unroll
        for (int v = 0; v < 8; ++v) { c0[v] = bv0; c1[v] = bv1; c2[v] = bv2; }

        #pragma unroll 4
        for (int k = 0; k < D_IN; k += 4) {
            v2f a  = *(const v2f*)(xr + k);
            v2f b0 = *(const v2f*)(w0 + k);
            v2f b1 = *(const v2f*)(w1 + k);
            v2f b2 = *(const v2f*)(w2 + k);
            c0 = wmma_f32(a, b0, c0);
            c1 = wmma_f32(a, b1, c1);
            c2 = wmma_f32(a, b2, c2);
        }
        #pragma unroll
        for (int v = 0; v < 8; ++v) {
            const int m = v + 8 * hi;
            float p;
            p = c0[v]; hs[m][j0_0 + ln] = (p > 0.f) ? p * p : 0.f;
            p = c1[v]; hs[m][j0_1 + ln] = (p > 0.f) ? p * p : 0.f;
            p = c2[v]; hs[m][j0_2 + ln] = (p > 0.f) ? p * p : 0.f;
        }
    }
    __syncthreads();

    // ================= scan: 383 sequential steps =================
    // Step i: pre = h @ W_i^T + b_i  over cols k<=i (rounded up to 4; extra cols
    // hit pre-zeroed W so contribute 0), rows j>i (tiles below start are all-zero
    // W/b so sqrelu(0)=0 leaves h unchanged -> no explicit masks, no mid-step barrier).
    for (int i = 0; i < H - 1; ++i) {
        const int kend  = ((i + 4) >> 2) << 2;     // ceil((i+1)/4)*4
        const int jt0   = ((i + 1) >> 4) + wv;     // first j-tile for this wave
        const int j0_0  = jt0 * 16, j0_1 = j0_0 + 128, j0_2 = j0_0 + 256;
        const bool act0 = jt0 < 24, act1 = (jt0 + 8) < 24, act2 = (jt0 + 16) < 24;

        const float* wbase = W_pad + (size_t)i * H * H;
        const float* brow  = b_pad + (size_t)i * H;
        const float* w0 = wbase + (size_t)(j0_0 + ln) * H + 2 * hi;
        const float* w1 = wbase + (size_t)(j0_1 + ln) * H + 2 * hi;
        const float* w2 = wbase + (size_t)(j0_2 + ln) * H + 2 * hi;

        v8f c0 = {}, c1 = {}, c2 = {};
        if (act0) { const float bv = brow[j0_0 + ln];
            #pragma unroll
            for (int v = 0; v < 8; ++v) c0[v] = bv; }
        if (act1) { const float bv = brow[j0_1 + ln];
            #pragma unroll
            for (int v = 0; v < 8; ++v) c1[v] = bv; }
        if (act2) { const float bv = brow[j0_2 + ln];
            #pragma unroll
            for (int v = 0; v < 8; ++v) c2[v] = bv; }

        for (int k = 0; k < kend; k += 4) {
            v2f a = *(const v2f*)&hs[ln][k + 2 * hi];
            if (act0) { v2f b = *(const v2f*)(w0 + k); c0 = wmma_f32(a, b, c0); }
            if (act1) { v2f b = *(const v2f*)(w1 + k); c1 = wmma_f32(a, b, c1); }
            if (act2) { v2f b = *(const v2f*)(w2 + k); c2 = wmma_f32(a, b, c2); }
        }

        if (act0) {
            #pragma unroll
            for (int v = 0; v < 8; ++v) { float p = c0[v]; if (p > 0.f) hs[v + 8*hi][j0_0 + ln] += p * p; }
        }
        if (act1) {
            #pragma unroll
            for (int v = 0; v < 8; ++v) { float p = c1[v]; if (p > 0.f) hs[v + 8*hi][j0_1 + ln] += p * p; }
        }
        if (act2) {
            #pragma unroll
            for (int v = 0; v < 8; ++v) { float p = c2[v]; if (p > 0.f) hs[v + 8*hi][j0_2 + ln] += p * p; }
        }
        __syncthreads();
    }

    // ================= output layer: out = h @ Wo^T + bo =================
    for (int t = 0; t < 8; ++t) {
        const int j0 = (wv + 8 * t) * 16;
        const float* wo = Wo + (size_t)(j0 + ln) * H + 2 * hi;
        const float bv = bo[j0 + ln];
        v8f c;
        #pragma unroll
        for (int v = 0; v < 8; ++v) c[v] = bv;

        #pragma unroll 4
        for (int k = 0; k < H; k += 4) {
            v2f a = *(const v2f*)&hs[ln][k + 2 * hi];
            v2f b = *(const v2f*)(wo + k);
            c = wmma_f32(a, b, c);
        }
        #pragma unroll
        for (int v = 0; v < 8; ++v)
            out[(size_t)(rowbase + v + 8 * hi) * D_OUT + j0 + ln] = c[v];
    }
}

extern "C" void kernel_launch(void* const* d_in, const int* in_sizes, int n_in,
                              void* d_out, int out_size, void* d_ws, size_t ws_size,
                              hipStream_t stream) {
    const float* x     = (const float*)d_in[0];
    const float* Wi    = (const float*)d_in[1];
    const float* bi    = (const float*)d_in[2];
    const float* W_pad = (const float*)d_in[3];
    const float* b_pad = (const float*)d_in[4];
    const float* Wo    = (const float*)d_in[5];
    const float* bo    = (const float*)d_in[6];
    float* out = (float*)d_out;

    freenet_kernel<<<B_TOT / BT, 256, 0, stream>>>(x, Wi, bi, W_pad, b_pad, Wo, bo, out);
}